// GetKnnGraph_28475633173130
// MI455X (gfx1250) — compile-verified
//
#include <hip/hip_runtime.h>

#define BATCH 8
#define NPTS  4096
#define CH    64
#define KNN   20
#define ROWS_PER_BLOCK 128
#define JT    16
#define LDS_STRIDE 17   // pad to avoid 16-float-stride bank conflicts on 64 banks

typedef __attribute__((ext_vector_type(2))) float v2f;
typedef __attribute__((ext_vector_type(8))) float v8f;

// ---------------------------------------------------------------------------
// Kernel 1: sq[b][n] = sum_c x[b][c][n]^2   (x is (B, C, N) contiguous)
// ---------------------------------------------------------------------------
__global__ __launch_bounds__(256) void knn_sqnorm_kernel(
    const float* __restrict__ x, float* __restrict__ sq) {
  int gid = blockIdx.x * 256 + threadIdx.x;          // over B*N
  int b = gid / NPTS;
  int n = gid - b * NPTS;
  const float* xb = x + (size_t)b * CH * NPTS + n;
  float s = 0.0f;
#pragma unroll
  for (int c = 0; c < CH; ++c) {
    float v = xb[(size_t)c * NPTS];                  // lanes: consecutive n -> coalesced
    s = fmaf(v, v, s);
  }
  sq[gid] = s;
}

// ---------------------------------------------------------------------------
// Kernel 2: fused WMMA gram tiles + per-row top-K selection.
// Block = 128 threads = 4 waves, owns 128 rows of one batch.
// Wave w holds A fragments for rows [row0+32w, row0+32w+32) (two 16-row tiles),
// loops over 256 column tiles of 16. Each tile's C=64 reduction is split into
// even/odd K partial accumulators -> 4 independent v_wmma_f32_16x16x4_f32
// chains per wave (dependency distance 4) to keep the XDL pipe full.
// ---------------------------------------------------------------------------
__global__ __launch_bounds__(128) void knn_topk_kernel(
    const float* __restrict__ x, const float* __restrict__ sq,
    int* __restrict__ out) {
  __shared__ float tile[ROWS_PER_BLOCK * LDS_STRIDE];

  const int b    = blockIdx.x >> 5;                  // 32 row-chunks per batch
  const int row0 = (blockIdx.x & 31) * ROWS_PER_BLOCK;
  const int lane = threadIdx.x & 31;
  const int wave = threadIdx.x >> 5;
  const int half = lane >> 4;                        // lane-group 0/1
  const int l16  = lane & 15;

  const float* xb  = x  + (size_t)b * CH * NPTS;
  const float* sqb = sq + (size_t)b * NPTS;

  // A fragments resident across the whole j loop.
  // A 16x4 f32 layout: lane L -> row M = L%16; VGPR r -> K = r + 2*(L/16).
  v2f afrag[2][16];
#pragma unroll
  for (int t = 0; t < 2; ++t) {
    const int m = row0 + wave * 32 + t * 16 + l16;
#pragma unroll
    for (int s = 0; s < 16; ++s) {
      const int k = s * 4 + half * 2;
      afrag[t][s].x = xb[(size_t)k       * NPTS + m];
      afrag[t][s].y = xb[(size_t)(k + 1) * NPTS + m];
    }
  }

  // Register-resident sorted top-K (ascending key = ||j||^2 - 2*dot).
  float bestd[KNN];
  int   besti[KNN];
#pragma unroll
  for (int q = 0; q < KNN; ++q) { bestd[q] = 3.4e38f; besti[q] = 0; }
  const int myrow = row0 + (int)threadIdx.x;         // row this thread selects for

  for (int j0 = 0; j0 < NPTS; j0 += JT) {
    v8f acc0e = {}, acc0o = {};
    v8f acc1e = {}, acc1o = {};
#pragma unroll
    for (int s = 0; s < 8; ++s) {
      const int ke = (2 * s)     * 4 + half * 2;
      const int ko = (2 * s + 1) * 4 + half * 2;
      v2f bfe, bfo;                                  // B 4x16: col = L%16, K = r + 2*(L/16)
      bfe.x = xb[(size_t)ke       * NPTS + (j0 + l16)];
      bfe.y = xb[(size_t)(ke + 1) * NPTS + (j0 + l16)];
      bfo.x = xb[(size_t)ko       * NPTS + (j0 + l16)];
      bfo.y = xb[(size_t)(ko + 1) * NPTS + (j0 + l16)];
      acc0e = __builtin_amdgcn_wmma_f32_16x16x4_f32(
          false, afrag[0][2 * s],     false, bfe, (short)0, acc0e, false, false);
      acc1e = __builtin_amdgcn_wmma_f32_16x16x4_f32(
          false, afrag[1][2 * s],     false, bfe, (short)0, acc1e, false, false);
      acc0o = __builtin_amdgcn_wmma_f32_16x16x4_f32(
          false, afrag[0][2 * s + 1], false, bfo, (short)0, acc0o, false, false);
      acc1o = __builtin_amdgcn_wmma_f32_16x16x4_f32(
          false, afrag[1][2 * s + 1], false, bfo, (short)0, acc1o, false, false);
    }

    // Fold ||j||^2 at the store; C/D layout: VGPR r, lanes>=16 -> M = r+8.
    const float sj = sqb[j0 + l16];
    const int rbase = wave * 32 + half * 8;
#pragma unroll
    for (int r = 0; r < 8; ++r) {
      tile[(rbase + r)      * LDS_STRIDE + l16] = fmaf(-2.0f, acc0e[r] + acc0o[r], sj);
      tile[(rbase + 16 + r) * LDS_STRIDE + l16] = fmaf(-2.0f, acc1e[r] + acc1o[r], sj);
    }
    __syncthreads();

    // Each thread scans its own row's 16 candidates.
#pragma unroll
    for (int jj = 0; jj < JT; ++jj) {
      const float d = tile[threadIdx.x * LDS_STRIDE + jj];
      const int   j = j0 + jj;
      if (j != myrow && d < bestd[KNN - 1]) {        // loop=False: exclude self
        bestd[KNN - 1] = d;
        besti[KNN - 1] = j;
#pragma unroll
        for (int q = KNN - 1; q > 0; --q) {          // one bubble pass re-sorts
          if (bestd[q] < bestd[q - 1]) {
            float td = bestd[q]; bestd[q] = bestd[q - 1]; bestd[q - 1] = td;
            int   ti = besti[q]; besti[q] = besti[q - 1]; besti[q - 1] = ti;
          }
        }
      }
    }
    __syncthreads();
  }

  // edge_index shape (2, B, N, K): [0]=source(neighbor), [1]=target(center).
  const size_t o0 = ((size_t)b * NPTS + myrow) * KNN;
  const size_t o1 = (size_t)BATCH * NPTS * KNN + o0;
#pragma unroll
  for (int q = 0; q < KNN; ++q) {
    out[o0 + q] = besti[q];
    out[o1 + q] = myrow;
  }
}

// ---------------------------------------------------------------------------
extern "C" void kernel_launch(void* const* d_in, const int* in_sizes, int n_in,
                              void* d_out, int out_size, void* d_ws, size_t ws_size,
                              hipStream_t stream) {
  (void)in_sizes; (void)n_in; (void)out_size; (void)ws_size;
  const float* x  = (const float*)d_in[0];
  float*       sq = (float*)d_ws;                    // B*N floats = 128 KB scratch
  int*         out = (int*)d_out;

  knn_sqnorm_kernel<<<(BATCH * NPTS) / 256, 256, 0, stream>>>(x, sq);
  knn_topk_kernel<<<BATCH * (NPTS / ROWS_PER_BLOCK), ROWS_PER_BLOCK, 0, stream>>>(x, sq, out);
}